// GalaxyNetNormalizeOutput_35029753266648
// MI455X (gfx1250) — compile-verified
//
#include <hip/hip_runtime.h>
#include <hip/hip_bf16.h>

#define NCOLS 37
#define ROWS_PER_BLOCK 256

// ---------------------------------------------------------------------------
// gfx1250 async Global <-> LDS copies (ASYNCcnt-tracked, CDNA5 ISA + 07)
// LDS operand is the wave-relative byte offset (low 32 bits of the flat addr).
// ---------------------------------------------------------------------------
__device__ __forceinline__ uint32_t lds_off(const void* p) {
    return (uint32_t)(uintptr_t)p;   // addrspace(3) offset lives in addr[31:0]
}

__device__ __forceinline__ void async_load_b128(uint32_t lds, const float* g) {
    asm volatile("global_load_async_to_lds_b128 %0, %1, off"
                 :: "v"(lds), "v"(g) : "memory");
}
__device__ __forceinline__ void async_load_b32(uint32_t lds, const float* g) {
    asm volatile("global_load_async_to_lds_b32 %0, %1, off"
                 :: "v"(lds), "v"(g) : "memory");
}
__device__ __forceinline__ void async_store_b128(float* g, uint32_t lds) {
    asm volatile("global_store_async_from_lds_b128 %0, %1, off"
                 :: "v"(g), "v"(lds) : "memory");
}
__device__ __forceinline__ void async_store_b32(float* g, uint32_t lds) {
    asm volatile("global_store_async_from_lds_b32 %0, %1, off"
                 :: "v"(g), "v"(lds) : "memory");
}
__device__ __forceinline__ void wait_async0() {
    asm volatile("s_wait_asynccnt 0" ::: "memory");
}

// ---------------------------------------------------------------------------
// One normalization step on a register-resident row.
//  seg = v[S:E]; s = sum(seg); if (s>0) seg /= s; seg *= scale (scale==1 if none)
//  Scale indices are disjoint from [S,E) in every step, matching reference.
// ---------------------------------------------------------------------------
template <int S, int E, bool HAS_SCALE>
__device__ __forceinline__ void norm_seg(float (&v)[NCOLS], float scale) {
    float s = 0.f;
#pragma unroll
    for (int i = S; i < E; ++i) s += v[i];
    if (s > 0.f) {
        float inv = 1.0f / s;
#pragma unroll
        for (int i = S; i < E; ++i) v[i] *= inv;
    }
    if (HAS_SCALE) {
#pragma unroll
        for (int i = S; i < E; ++i) v[i] *= scale;
    }
}

__device__ __forceinline__ void process_row(float (&v)[NCOLS]) {
    norm_seg< 0,  3, false>(v, 1.f);
    norm_seg<15, 18, true >(v, v[0]);
    norm_seg< 3,  5, true >(v, v[1]);
    norm_seg<25, 28, true >(v, v[3]);
    norm_seg< 5,  7, true >(v, v[4]);
    norm_seg< 7,  9, true >(v, v[5] + v[6]);
    norm_seg<28, 31, true >(v, v[7]);
    norm_seg<31, 37, true >(v, v[28] + v[29] + v[30]);
    norm_seg< 9, 13, true >(v, v[8] + v[31] + v[32] + v[33] + v[34] + v[35] + v[36]);
    norm_seg<13, 15, false>(v, 1.f);
    norm_seg<18, 25, true >(v, v[13]);
}

// ---------------------------------------------------------------------------
// Kernel: tile rows through LDS for fully coalesced 128b global traffic.
// Row stride in LDS = 37 dwords (odd) -> bank-conflict-free per-lane access.
// ---------------------------------------------------------------------------
__global__ __launch_bounds__(ROWS_PER_BLOCK)
void galaxy_normalize_kernel(const float* __restrict__ x,
                             float* __restrict__ out,
                             int nrows) {
    __shared__ __align__(16) float tile[ROWS_PER_BLOCK * NCOLS];

    const long long rowBase = (long long)blockIdx.x * ROWS_PER_BLOCK;
    const int validRows = min(ROWS_PER_BLOCK, (int)(nrows - rowBase));
    const int nFloats   = validRows * NCOLS;          // floats this block owns
    const int n4        = nFloats >> 2;               // full float4 chunks
    const float* gin  = x   + rowBase * NCOLS;        // block base: 16B aligned
    float*       gout = out + rowBase * NCOLS;        // (256*148 % 16 == 0)
    const uint32_t ldsBase = lds_off(&tile[0]);
    const int tid = threadIdx.x;

    // -------- Phase 1: async global -> LDS (coalesced b128 + b32 tail) -----
    for (int i = tid; i < n4; i += ROWS_PER_BLOCK)
        async_load_b128(ldsBase + (uint32_t)i * 16u, gin + i * 4);
    for (int i = (n4 << 2) + tid; i < nFloats; i += ROWS_PER_BLOCK)
        async_load_b32(ldsBase + (uint32_t)i * 4u, gin + i);
    wait_async0();
    __syncthreads();

    // -------- Phase 2: one row per lane, all math in registers -------------
    if (tid < validRows) {
        float v[NCOLS];
        const int base = tid * NCOLS;
#pragma unroll
        for (int i = 0; i < NCOLS; ++i) v[i] = tile[base + i];
        process_row(v);
#pragma unroll
        for (int i = 0; i < NCOLS; ++i) tile[base + i] = v[i];
    }
    __syncthreads();

    // -------- Phase 3: async LDS -> global (coalesced) ---------------------
    for (int i = tid; i < n4; i += ROWS_PER_BLOCK)
        async_store_b128(gout + i * 4, ldsBase + (uint32_t)i * 16u);
    for (int i = (n4 << 2) + tid; i < nFloats; i += ROWS_PER_BLOCK)
        async_store_b32(gout + i, ldsBase + (uint32_t)i * 4u);
    // S_ENDPGM performs an implicit wait-idle (ASYNCcnt drains before retire).
}

// ---------------------------------------------------------------------------
extern "C" void kernel_launch(void* const* d_in, const int* in_sizes, int n_in,
                              void* d_out, int out_size, void* d_ws, size_t ws_size,
                              hipStream_t stream) {
    const float* x = (const float*)d_in[0];
    float* out = (float*)d_out;
    const int nrows = in_sizes[0] / NCOLS;            // 2,000,000
    const int blocks = (nrows + ROWS_PER_BLOCK - 1) / ROWS_PER_BLOCK;
    galaxy_normalize_kernel<<<blocks, ROWS_PER_BLOCK, 0, stream>>>(x, out, nrows);
}